// TensorInteraction_9216999817544
// MI455X (gfx1250) — compile-verified
//
#include <hip/hip_runtime.h>

typedef float v2f __attribute__((ext_vector_type(2)));
typedef float v8f __attribute__((ext_vector_type(8)));

#define F_DIM 128   // feature dim (== A)
#define NB_DIM 50   // radial basis count

// ---------------------------------------------------------------------------
// CDNA5 fp32 WMMA: D(16x16,f32) = A(16x4,f32) * B(4x16,f32) + C
// ---------------------------------------------------------------------------
__device__ __forceinline__ v8f wmma_f32_k4(v2f a, v2f b, v8f c) {
  // 8 args: (neg_a, A, neg_b, B, c_mod, C, reuse_a, reuse_b)
  return __builtin_amdgcn_wmma_f32_16x16x4_f32(false, a, false, b,
                                               (short)0, c, false, false);
}

__device__ __forceinline__ float fast_rcp(float x) {
  return __builtin_amdgcn_rcpf(x);            // v_rcp_f32
}

__device__ __forceinline__ float swish_f(float x) {
  return x * fast_rcp(1.0f + __expf(-x));     // no IEEE div
}

__device__ __forceinline__ float moll_cutoff(float r) {
  float mask = (r + 1e-7f < 5.0f) ? 1.0f : 0.0f;
  float rm = r * mask * 0.2f;                 // r*mask/cutoff
  float ex = 1.0f - fast_rcp(1.0f - rm * rm);
  return __expf(ex) * mask;
}

__device__ __forceinline__ void atomic_add_f32(float* p, float v) {
  __hip_atomic_fetch_add(p, v, __ATOMIC_RELAXED, __HIP_MEMORY_SCOPE_AGENT);
}

// ---------------------------------------------------------------------------
// Kernel 1: dipole normalization, written TRANSPOSED as mu_n[N][3][F]
//   mu_n[n][i][f] = mu[n][f][i] / (coef[f]^2 * ||mu[n][f]+eps|| + 1)
// ---------------------------------------------------------------------------
__global__ void prep_mu_kernel(const float* __restrict__ mu,
                               const float* __restrict__ coef,
                               float* __restrict__ mu_n, int NF) {
  int idx = blockIdx.x * blockDim.x + threadIdx.x;
  if (idx >= NF) return;
  int n = idx >> 7;
  int f = idx & (F_DIM - 1);
  float x = mu[idx * 3 + 0], y = mu[idx * 3 + 1], z = mu[idx * 3 + 2];
  float xe = x + 1e-7f, ye = y + 1e-7f, ze = z + 1e-7f;
  float nrm = sqrtf(xe * xe + ye * ye + ze * ze);
  float cf = coef[f];
  float inv = fast_rcp(cf * cf * nrm + 1.0f);
  float* o = mu_n + (size_t)n * (3 * F_DIM) + f;
  o[0 * F_DIM] = x * inv;
  o[1 * F_DIM] = y * inv;
  o[2 * F_DIM] = z * inv;
}

// ---------------------------------------------------------------------------
// Kernel 2/4: Y = swish(X @ W1^T + b1) @ W2^T + b2, all dims 128.
// One block = 16 rows; 8 waves each own one 16-col tile; WMMA f32 16x16x4.
// ---------------------------------------------------------------------------
__global__ void mlp2_128_kernel(const float* __restrict__ X,
                                const float* __restrict__ W1,
                                const float* __restrict__ b1,
                                const float* __restrict__ W2,
                                const float* __restrict__ b2,
                                float* __restrict__ Y, int n_rows) {
  __shared__ float sX[16][129];
  __shared__ float sH[16][129];
  const int tid = threadIdx.x;
  const int r0 = blockIdx.x * 16;

  for (int p = tid; p < 16 * F_DIM; p += 256) {
    int r = p >> 7, c = p & (F_DIM - 1);
    int gr = r0 + r;
    sX[r][c] = (gr < n_rows) ? X[(size_t)gr * F_DIM + c] : 0.0f;
  }
  __syncthreads();

  const int lane = tid & 31;
  const int wv = tid >> 5;
  const int lrow = lane & 15;
  const int kh = (lane >> 4) << 1;          // 0 or 2
  const int rbase = (lane >> 4) << 3;       // 0 or 8
  const int colg = wv * 16 + lrow;

  // GEMM 1: hidden = X @ W1^T
  v8f acc = {};
  #pragma unroll 8
  for (int k0 = 0; k0 < F_DIM; k0 += 4) {
    int k1 = k0 + kh;
    v2f a, b;
    a.x = sX[lrow][k1];            a.y = sX[lrow][k1 + 1];
    b.x = W1[colg * F_DIM + k1];   b.y = W1[colg * F_DIM + k1 + 1];
    acc = wmma_f32_k4(a, b, acc);
  }
  float bc1 = b1[colg];
  #pragma unroll
  for (int r = 0; r < 8; ++r) {
    int row = r + rbase;
    sH[row][colg] = swish_f(acc[r] + bc1);
  }
  __syncthreads();

  // GEMM 2: Y = hidden @ W2^T
  v8f acc2 = {};
  #pragma unroll 8
  for (int k0 = 0; k0 < F_DIM; k0 += 4) {
    int k1 = k0 + kh;
    v2f a, b;
    a.x = sH[lrow][k1];            a.y = sH[lrow][k1 + 1];
    b.x = W2[colg * F_DIM + k1];   b.y = W2[colg * F_DIM + k1 + 1];
    acc2 = wmma_f32_k4(a, b, acc2);
  }
  float bc2 = b2[colg];
  #pragma unroll
  for (int r = 0; r < 8; ++r) {
    int row = r + rbase;
    int gr = r0 + row;
    if (gr < n_rows) Y[(size_t)gr * F_DIM + colg] = acc2[r] + bc2;
  }
}

// ---------------------------------------------------------------------------
// Kernel 3: fused edge phase. 16 edges / block (256 thr = 8 waves).
//   radial = swish(fij @ We1^T + be1) @ We2^T + be2, scaled by cutoff/r^5
//   then geometry + atomic scatter into vn_acc[N,F] and mvt_acc[N,3,F].
// ---------------------------------------------------------------------------
__global__ void edge_kernel(const float* __restrict__ rij,
                            const float* __restrict__ vij,
                            const float* __restrict__ fij,
                            const int* __restrict__ src,
                            const int* __restrict__ dst,
                            const float* __restrict__ We1,
                            const float* __restrict__ be1,
                            const float* __restrict__ We2,
                            const float* __restrict__ be2,
                            const float* __restrict__ mu_n,
                            const float* __restrict__ q,
                            float* __restrict__ vn_acc,
                            float* __restrict__ mvt_acc, int E) {
  __shared__ float sF[16][53];    // f_ij tile, K padded 50 -> 52
  __shared__ float sH[16][129];   // hidden, then reused for radial
  __shared__ float sR[16];
  __shared__ float sV[16][3];
  __shared__ int   sS[16];
  __shared__ int   sD[16];

  const int tid = threadIdx.x;
  const int e0 = blockIdx.x * 16;

  // prefetch next tile's f_ij rows (streamed HBM input): 16*50*4B = 3200B
  if ((size_t)(e0 + 32) * NB_DIM <= (size_t)E * NB_DIM && tid < 13) {
    const char* nxt = (const char*)(fij + (size_t)(e0 + 16) * NB_DIM);
    __builtin_prefetch(nxt + tid * 256, 0, 3);   // global_prefetch_b8
  }

  for (int p = tid; p < 16 * 52; p += 256) {
    int r = p / 52, k = p - 52 * r;
    int e = e0 + r;
    sF[r][k] = (k < NB_DIM && e < E) ? fij[(size_t)e * NB_DIM + k] : 0.0f;
  }
  if (tid < 16) {
    int e = e0 + tid; if (e >= E) e = E - 1;
    sR[tid] = rij[e]; sS[tid] = src[e]; sD[tid] = dst[e];
  }
  if (tid < 48) {
    int r = tid / 3, c = tid - 3 * r;
    int e = e0 + r; if (e >= E) e = E - 1;
    sV[r][c] = vij[(size_t)e * 3 + c];
  }
  __syncthreads();

  const int lane = tid & 31;
  const int wv = tid >> 5;
  const int lrow = lane & 15;
  const int kh = (lane >> 4) << 1;
  const int rbase = (lane >> 4) << 3;
  const int colg = wv * 16 + lrow;

  // GEMM 1: hidden[16,128] = fij[16,52] @ We1^T (zeros pad K 50->52)
  v8f acc = {};
  #pragma unroll
  for (int k0 = 0; k0 < 52; k0 += 4) {
    int k1 = k0 + kh, k2 = k0 + kh + 1;
    v2f a, b;
    a.x = sF[lrow][k1];  a.y = sF[lrow][k2];
    b.x = (k1 < NB_DIM) ? We1[colg * NB_DIM + k1] : 0.0f;
    b.y = (k2 < NB_DIM) ? We1[colg * NB_DIM + k2] : 0.0f;
    acc = wmma_f32_k4(a, b, acc);
  }
  float bc1 = be1[colg];
  #pragma unroll
  for (int r = 0; r < 8; ++r) sH[r + rbase][colg] = swish_f(acc[r] + bc1);
  __syncthreads();

  // GEMM 2: radial[16,128] = hidden @ We2^T
  v8f acc2 = {};
  #pragma unroll 8
  for (int k0 = 0; k0 < F_DIM; k0 += 4) {
    int k1 = k0 + kh;
    v2f a, b;
    a.x = sH[lrow][k1];            a.y = sH[lrow][k1 + 1];
    b.x = We2[colg * F_DIM + k1];  b.y = We2[colg * F_DIM + k1 + 1];
    acc2 = wmma_f32_k4(a, b, acc2);
  }
  __syncthreads();   // everyone done reading hidden before overwrite

  float bc2 = be2[colg];
  #pragma unroll
  for (int r = 0; r < 8; ++r) {
    int row = r + rbase;
    float rr = sR[row];
    float r5 = rr * rr; r5 = r5 * r5 * rr;
    float scale = moll_cutoff(rr) * fast_rcp(r5 + 1e-10f);
    sH[row][colg] = (acc2[r] + bc2) * scale;   // radial, cutoff-smoothed
  }
  __syncthreads();

  // geometry + scatter: 16 edges x 128 features; consecutive lanes get
  // consecutive f for a fixed edge row -> all global accesses coalesced.
  for (int p = tid; p < 16 * F_DIM; p += 256) {
    int r = p >> 7, f = p & (F_DIM - 1);
    int e = e0 + r;
    if (e >= E) continue;
    int s = sS[r], d = sD[r];
    float rr = sR[r], r2 = rr * rr;
    float vx = sV[r][0], vy = sV[r][1], vz = sV[r][2];
    const float* pu = mu_n + (size_t)s * (3 * F_DIM) + f;   // [N,3,F]
    const float* pv = mu_n + (size_t)d * (3 * F_DIM) + f;
    float ux = pu[0 * F_DIM], uy = pu[1 * F_DIM], uz = pu[2 * F_DIM];
    float wx = pv[0 * F_DIM], wy = pv[1 * F_DIM], wz = pv[2 * F_DIM];
    float duv = ux * wx + uy * wy + uz * wz;
    float o1 = ux * vx + uy * vy + uz * vz;
    float o2 = wx * vx + wy * vy + wz * vz;
    float rad = sH[r][f];
    // scalar channel: (mu_u.mu_v * r^2 - 3 o1 o2) * radial
    float ve = (duv * r2 - 3.0f * o1 * o2) * rad;
    atomic_add_f32(&vn_acc[(size_t)d * F_DIM + f], ve);
    // vector channel: T mu = r^2 mu - 3 v (v.mu)
    float qs = q[(size_t)s * F_DIM + f];
    float qd = q[(size_t)d * F_DIM + f];
    float mvx = ((r2 * wx - 3.0f * vx * o2) * qs - (r2 * ux - 3.0f * vx * o1) * qd) * rad;
    float mvy = ((r2 * wy - 3.0f * vy * o2) * qs - (r2 * uy - 3.0f * vy * o1) * qd) * rad;
    float mvz = ((r2 * wz - 3.0f * vz * o2) * qs - (r2 * uz - 3.0f * vz * o1) * qd) * rad;
    float* pm = mvt_acc + (size_t)d * (3 * F_DIM) + f;       // [N,3,F]
    atomic_add_f32(pm + 0 * F_DIM, mvx);
    atomic_add_f32(pm + 1 * F_DIM, mvy);
    atomic_add_f32(pm + 2 * F_DIM, mvz);
  }
}

// ---------------------------------------------------------------------------
// Kernel 5: mvt[n,a,i] = sum_f mvt_acc[n,i,f] * Wv[a,f]
// GEMM over flattened rows r = n*3+i (M = 3N), K = F, cols = A.
// ---------------------------------------------------------------------------
__global__ void mvt_kernel(const float* __restrict__ mvt_acc,
                           const float* __restrict__ Wv,
                           float* __restrict__ out, int n_rows3) {
  __shared__ float sA[16][129];
  const int tid = threadIdx.x;
  const int r0 = blockIdx.x * 16;

  for (int p = tid; p < 16 * F_DIM; p += 256) {
    int r = p >> 7, f = p & (F_DIM - 1);
    int gr = r0 + r;
    if (gr < n_rows3) {
      int n = gr / 3, i = gr - 3 * n;
      sA[r][f] = mvt_acc[(size_t)n * (3 * F_DIM) + i * F_DIM + f];  // coalesced
    } else {
      sA[r][f] = 0.0f;
    }
  }
  __syncthreads();

  const int lane = tid & 31;
  const int wv = tid >> 5;
  const int lrow = lane & 15;
  const int kh = (lane >> 4) << 1;
  const int rbase = (lane >> 4) << 3;
  const int colg = wv * 16 + lrow;   // output channel a

  v8f acc = {};
  #pragma unroll 8
  for (int k0 = 0; k0 < F_DIM; k0 += 4) {
    int k1 = k0 + kh;
    v2f a, b;
    a.x = sA[lrow][k1];            a.y = sA[lrow][k1 + 1];
    b.x = Wv[colg * F_DIM + k1];   b.y = Wv[colg * F_DIM + k1 + 1];
    acc = wmma_f32_k4(a, b, acc);
  }
  #pragma unroll
  for (int r = 0; r < 8; ++r) {
    int row = r + rbase;
    int gr = r0 + row;
    if (gr < n_rows3) {
      int n = gr / 3, i = gr - 3 * n;
      out[(size_t)n * (F_DIM * 3) + colg * 3 + i] = acc[r];  // [N,A,3] output
    }
  }
}

// ---------------------------------------------------------------------------
extern "C" void kernel_launch(void* const* d_in, const int* in_sizes, int n_in,
                              void* d_out, int out_size, void* d_ws, size_t ws_size,
                              hipStream_t stream) {
  (void)n_in; (void)out_size; (void)ws_size;
  const float* mu   = (const float*)d_in[0];
  const float* rij  = (const float*)d_in[1];
  const float* vij  = (const float*)d_in[2];
  const float* fij  = (const float*)d_in[3];
  const float* feat = (const float*)d_in[4];
  const int*   src  = (const int*)d_in[5];
  const int*   dst  = (const int*)d_in[6];
  const float* coef = (const float*)d_in[7];
  const float* Wq1  = (const float*)d_in[8];
  const float* bq1  = (const float*)d_in[9];
  const float* Wq2  = (const float*)d_in[10];
  const float* bq2  = (const float*)d_in[11];
  const float* Wv   = (const float*)d_in[12];
  const float* Wd1  = (const float*)d_in[13];
  const float* bd1  = (const float*)d_in[14];
  const float* Wd2  = (const float*)d_in[15];
  const float* bd2  = (const float*)d_in[16];
  const float* We1  = (const float*)d_in[17];
  const float* be1  = (const float*)d_in[18];
  const float* We2  = (const float*)d_in[19];
  const float* be2  = (const float*)d_in[20];

  const int E = in_sizes[1];
  const int N = in_sizes[4] / F_DIM;

  // workspace (floats): mu_n[N*3*F] | q[N*F] | vn_acc[N*F] | mvt_acc[N*3*F]
  float* ws      = (float*)d_ws;
  float* mu_n    = ws;
  float* qbuf    = mu_n + (size_t)N * F_DIM * 3;
  float* vn_acc  = qbuf + (size_t)N * F_DIM;
  float* mvt_acc = vn_acc + (size_t)N * F_DIM;

  // zero the atomic accumulators (vn_acc and mvt_acc are adjacent)
  hipMemsetAsync(vn_acc, 0, (size_t)N * F_DIM * 4 * sizeof(float), stream);

  prep_mu_kernel<<<(N * F_DIM + 255) / 256, 256, 0, stream>>>(mu, coef, mu_n, N * F_DIM);

  mlp2_128_kernel<<<(N + 15) / 16, 256, 0, stream>>>(feat, Wq1, bq1, Wq2, bq2, qbuf, N);

  edge_kernel<<<(E + 15) / 16, 256, 0, stream>>>(rij, vij, fij, src, dst,
                                                 We1, be1, We2, be2,
                                                 mu_n, qbuf, vn_acc, mvt_acc, E);

  float* out_v   = (float*)d_out;                    // [N, A]
  float* out_mvt = out_v + (size_t)N * F_DIM;        // [N, A, 3]

  mlp2_128_kernel<<<(N + 15) / 16, 256, 0, stream>>>(vn_acc, Wd1, bd1, Wd2, bd2, out_v, N);

  mvt_kernel<<<(3 * N + 15) / 16, 256, 0, stream>>>(mvt_acc, Wv, out_mvt, 3 * N);
}